// VectorQuantizerEMA_5523327943135
// MI455X (gfx1250) — compile-verified
//
#include <hip/hip_runtime.h>

typedef __attribute__((ext_vector_type(2))) float v2f;
typedef __attribute__((ext_vector_type(8))) float v8f;

#define K_CODES 1024
#define D_DIM   64
#define HW      1024                 // 32*32
#define OUT_Q_ELEMS 4194304          // 64*64*32*32
#define ENC_OFF     4194304
#define LOSS_OFF    4259840
#define NLL_OFF     4259841
#define N_TILES     4096             // 65536 rows / 16
#define WAVES_PER_BLOCK 8

// ||e_k||^2 for all 1024 codebook rows.
__global__ void vq_esq_kernel(const float* __restrict__ emb, float* __restrict__ esq) {
    int k = blockIdx.x * blockDim.x + threadIdx.x;
    const float4* p = (const float4*)(emb + k * D_DIM);
    float s = 0.f;
#pragma unroll
    for (int i = 0; i < 16; ++i) {
        float4 v = p[i];
        s += v.x * v.x + v.y * v.y + v.z * v.z + v.w * v.w;
    }
    esq[k] = s;
}

__global__ __launch_bounds__(256) void vq_main_kernel(
        const float* __restrict__ x, const float* __restrict__ emb,
        const float* __restrict__ esq, float* __restrict__ out,
        float* __restrict__ wsum) {
    __shared__ int sK[WAVES_PER_BLOCK][16];
    const int lane = threadIdx.x & 31;
    const int wave = threadIdx.x >> 5;
    const int tile = blockIdx.x * WAVES_PER_BLOCK + wave;   // 0..4095
    const int n_base = tile << 4;                           // first row of tile
    const int b  = n_base >> 10;                            // batch index
    const int hw = n_base & (HW - 1);                       // spatial offset
    const int r  = lane & 15;                               // row (A) / col (B) in tile
    const int h  = lane >> 4;                               // lane-half -> K pair

    // A tile: 16 rows x 64 dims of X (NCHW: dim stride = HW floats).
    // fp32 16x4 A layout: lanes 0-15 rows, v0/v1 = K=2h / 2h+1 per 4-wide step.
    const float* xin = x + b * (D_DIM * HW) + hw + r;
    float a0[16], a1[16];
#pragma unroll
    for (int kk = 0; kk < 16; ++kk) {
        int c0 = kk * 4 + h * 2;
        a0[kk] = xin[c0 * HW];
        a1[kk] = xin[(c0 + 1) * HW];
    }

    float best[8];
    int   bidx[8];
#pragma unroll
    for (int j = 0; j < 8; ++j) { best[j] = 3.4e38f; bidx[j] = 0; }

    for (int t = 0; t < 64; ++t) {
        const int col = t * 16 + r;
        const float* erow = emb + col * D_DIM;               // row-major codebook
        if (t < 63) __builtin_prefetch(erow + 16 * D_DIM, 0, 0);
        v8f c = {};
#pragma unroll
        for (int kk = 0; kk < 16; ++kk) {
            v2f av; av.x = a0[kk]; av.y = a1[kk];
            v2f bv = *(const v2f*)(erow + kk * 4 + h * 2);    // aligned float2
            c = __builtin_amdgcn_wmma_f32_16x16x4_f32(false, av, false, bv,
                                                      (short)0, c, false, false);
        }
        // score = ||e||^2 - 2*dot  (||x||^2 constant per row, dropped)
        const float eq = esq[col];
#pragma unroll
        for (int j = 0; j < 8; ++j) {
            float d = fmaf(-2.0f, c[j], eq);
            if (d < best[j]) { best[j] = d; bidx[j] = col; }  // cols ascend with t
        }
    }

    // Argmin across the 16 lanes of each half; ties -> smaller index (jnp.argmin).
#pragma unroll
    for (int j = 0; j < 8; ++j) {
        float v = best[j]; int i = bidx[j];
#pragma unroll
        for (int m = 8; m >= 1; m >>= 1) {
            float v2 = __shfl_xor(v, m, 32);
            int   i2 = __shfl_xor(i, m, 32);
            if (v2 < v || (v2 == v && i2 < i)) { v = v2; i = i2; }
        }
        if (r == j) sK[wave][j + h * 8] = i;   // row = j + 8*half
    }
    __syncthreads();

    const int myk = sK[wave][r];
    if (lane < 16) out[ENC_OFF + n_base + lane] = (float)sK[wave][lane];

    // Quantized write-back (NCHW, coalesced over rows) + squared-error partial.
    const float* eqr  = emb + myk * D_DIM;
    float*       outp = out + b * (D_DIM * HW) + hw + r;
    const float* xp   = x   + b * (D_DIM * HW) + hw + r;
    float acc = 0.f;
#pragma unroll
    for (int cc = 0; cc < 32; ++cc) {
        int c = cc * 2 + h;
        float q  = eqr[c];
        float xv = xp[c * HW];
        outp[c * HW] = q;
        float dq = q - xv;
        acc = fmaf(dq, dq, acc);
    }
#pragma unroll
    for (int m = 16; m >= 1; m >>= 1) acc += __shfl_xor(acc, m, 32);
    if (lane == 0) wsum[tile] = acc;           // deterministic per-tile partial
}

__global__ void vq_finalize_kernel(const float* __restrict__ wsum, float* __restrict__ out) {
    __shared__ float s[256];
    float a = 0.f;
    for (int i = threadIdx.x; i < N_TILES; i += 256) a += wsum[i];
    s[threadIdx.x] = a;
    __syncthreads();
    for (int st = 128; st > 0; st >>= 1) {
        if ((int)threadIdx.x < st) s[threadIdx.x] += s[threadIdx.x + st];
        __syncthreads();
    }
    if (threadIdx.x == 0) {
        // loss = q_latent + 0.25*e_latent, both numerically mean((q-x)^2)
        out[LOSS_OFF] = 1.25f * s[0] / (float)OUT_Q_ELEMS;
        out[NLL_OFF]  = 1.0f;
    }
}

extern "C" void kernel_launch(void* const* d_in, const int* in_sizes, int n_in,
                              void* d_out, int out_size, void* d_ws, size_t ws_size,
                              hipStream_t stream) {
    const float* x   = (const float*)d_in[0];   // [64,64,32,32] f32
    const float* emb = (const float*)d_in[1];   // [1024,64] f32
    float* out  = (float*)d_out;
    float* ws   = (float*)d_ws;
    float* wsum = ws;                 // 4096 floats: per-tile loss partials
    float* esq  = ws + N_TILES;       // 1024 floats: ||e_k||^2

    vq_esq_kernel<<<K_CODES / 256, 256, 0, stream>>>(emb, esq);
    vq_main_kernel<<<N_TILES / WAVES_PER_BLOCK, 256, 0, stream>>>(x, emb, esq, out, wsum);
    vq_finalize_kernel<<<1, 256, 0, stream>>>(wsum, out);
}